// ROIPoolingLayer_48198122996307
// MI455X (gfx1250) — compile-verified
//
#include <hip/hip_runtime.h>
#include <hip/hip_bf16.h>

// Problem constants (match the reference).
#define PH 7
#define PW 7
#define BDIM 4
#define HDIM 56
#define WDIM 56
#define CDIM 256

// ROI max-pool, fp32.
//   features: [B,H,W,C]  rois: [R,5] (b, y1, x1, y2, x2) inclusive
//   out:      [R,PH,PW,C]
//
// Grid:  R*PH*PW blocks, one block per output bin.
// Block: 64 threads (2 wave32s); lane t owns channels [4t, 4t+4) -> float4.
// Bin extents from the reference's bin = floor(off*P/len) rule:
//   rows y in [y1 + ceil(ph*rh/PH), y1 + ceil((ph+1)*rh/PH))
//   cols x in [x1 + ceil(pw*rw/PW), x1 + ceil((pw+1)*rw/PW))
__global__ __launch_bounds__(64)
void roi_maxpool_kernel(const float* __restrict__ feat,
                        const int*   __restrict__ rois,
                        float*       __restrict__ out)
{
    const int idx = blockIdx.x;            // r * 49 + ph * 7 + pw
    const int r   = idx / (PH * PW);
    const int pb  = idx - r * (PH * PW);
    const int ph  = pb / PW;
    const int pw  = pb - ph * PW;

    const int* roi = rois + (size_t)r * 5;
    const int b  = roi[0];
    const int y1 = roi[1];
    const int x1 = roi[2];
    const int y2 = roi[3];
    const int x2 = roi[4];
    const int rh = y2 - y1 + 1;
    const int rw = x2 - x1 + 1;

    // ceil(a/b) for non-negative a, positive b
    const int ys = y1 + ( ph      * rh + PH - 1) / PH;
    const int ye = y1 + ((ph + 1) * rh + PH - 1) / PH;   // exclusive
    const int xs = x1 + ( pw      * rw + PW - 1) / PW;
    const int xe = x1 + ((pw + 1) * rw + PW - 1) / PW;   // exclusive

    const int c4 = threadIdx.x * 4;        // 64 lanes * 4 ch = 256 channels

    const float NEG_INF = -__builtin_inff();
    float4 acc = make_float4(NEG_INF, NEG_INF, NEG_INF, NEG_INF);

    const size_t imgBase = (size_t)b * HDIM * WDIM * CDIM;

    for (int y = ys; y < ye; ++y) {
        const float* rowp = feat + imgBase + ((size_t)y * WDIM) * CDIM + c4;

        // Prefetch the first cacheline of the next row's span (gfx1250
        // global_prefetch_b8). Cheap; helps pull the next row toward the WGP
        // while the current row's b128 loads are in flight.
        if (y + 1 < ye) {
            const float* nxt = feat + imgBase + ((size_t)(y + 1) * WDIM) * CDIM
                                    + (size_t)xs * CDIM + c4;
            __builtin_prefetch(nxt, 0, 0);
        }

        for (int x = xs; x < xe; ++x) {
            const float4 v = *reinterpret_cast<const float4*>(rowp + (size_t)x * CDIM);
            acc.x = fmaxf(acc.x, v.x);
            acc.y = fmaxf(acc.y, v.y);
            acc.z = fmaxf(acc.z, v.z);
            acc.w = fmaxf(acc.w, v.w);
        }
    }

    // Reference maps non-finite (empty-bin) results to 0. With roi sides >= 7
    // every bin is non-empty, but keep the clamp for exact parity.
    if (!(acc.x >= NEG_INF && acc.x <= __builtin_inff())) { /* never */ }
    acc.x = __builtin_isfinite(acc.x) ? acc.x : 0.0f;
    acc.y = __builtin_isfinite(acc.y) ? acc.y : 0.0f;
    acc.z = __builtin_isfinite(acc.z) ? acc.z : 0.0f;
    acc.w = __builtin_isfinite(acc.w) ? acc.w : 0.0f;

    float* op = out + ((size_t)idx) * CDIM + c4;
    *reinterpret_cast<float4*>(op) = acc;
}

extern "C" void kernel_launch(void* const* d_in, const int* in_sizes, int n_in,
                              void* d_out, int out_size, void* d_ws, size_t ws_size,
                              hipStream_t stream)
{
    const float* feat = reinterpret_cast<const float*>(d_in[0]);
    const int*   rois = reinterpret_cast<const int*>(d_in[1]);
    float*       out  = reinterpret_cast<float*>(d_out);

    const int R = in_sizes[1] / 5;                 // 128
    const int nblocks = R * PH * PW;               // 6272

    roi_maxpool_kernel<<<dim3(nblocks), dim3(64), 0, stream>>>(feat, rois, out);
}